// HoT_GNN_87385404604877
// MI455X (gfx1250) — compile-verified
//
#include <hip/hip_runtime.h>
#include <hip/hip_bf16.h>

// ---------------------------------------------------------------------------
// Sizes (fixed by the reference)
// ---------------------------------------------------------------------------
#define NN   4096
#define EE   8192
#define HN_  1024
#define KK_  8
#define NF_  128
#define EF_  64
#define NH1  256
#define NH2  256
#define EH_  64
#define EMB_ 258
#define BN_EPS 1e-5f

typedef __attribute__((ext_vector_type(2))) float v2f;
typedef __attribute__((ext_vector_type(4))) float v4f;
typedef __attribute__((ext_vector_type(8))) float v8f;

// ---------------------------------------------------------------------------
// Generic fp32 WMMA GEMM:  C[M,N] = act( A[M,K] @ op(B) + bias )
//   op(B) = B[K,N] row-major if TRANSB==0, else B stored [N,K] (i.e. X@W^T)
// Grid: gridDim.x = M/16.  Block: NWAVES waves, NWAVES = N/16.
// Each wave owns one 16x16 C tile. The 16xKT fp32 A tile is staged in LDS
// (A = the big matrix, each element touches HBM exactly once). B is skinny
// and L2-resident; fragments are loaded from a uniform SGPR base with 32-bit
// element offsets (SADDR form, cheap 32-bit offset bumps).
// WMMA fragment layouts per CDNA5 ISA 7.12.2:
//   A 16x4 f32 : v0 -> K=2*half, v1 -> K=2*half+1 ; M = lane&15
//   B 4x16 f32 : v0 -> K=2*half, v1 -> K=2*half+1 ; N = lane&15
//   C 16x16    : vgpr i -> M = i + 8*half ;         N = lane&15
// ---------------------------------------------------------------------------
#define KT 64

template<int NWAVES, int TRANSB, int HASBIAS, int RELU>
__global__ __launch_bounds__(NWAVES * 32)
void gemm_wmma_f32(const float* __restrict__ A, int lda,
                   const float* __restrict__ B, int ldb,
                   const float* __restrict__ bias,
                   float* __restrict__ C, int ldc, int K)
{
    constexpr int NT = NWAVES * 32;
    __shared__ float sA[16 * KT];

    const int tid  = threadIdx.x;
    const int lane = tid & 31;
    const int wave = tid >> 5;              // N-tile index within block
    const int row0 = blockIdx.x * 16;       // C row base
    const int half = lane >> 4;             // 0 | 1
    const int l16  = lane & 15;
    const int col  = wave * 16 + l16;       // this lane's C column

    v8f acc = {};

    // B fragment: 32-bit element offset from uniform base, constant stride.
    int boff, bstep;
    if (TRANSB) { boff = col * ldb + 2 * half;  bstep = 4; }
    else        { boff = (2 * half) * ldb + col; bstep = 4 * ldb; }

    // A cooperative-load offsets (all 32-bit; max index ~67M < 2^31)
    const int aoff0 = row0 * lda;

    for (int kb = 0; kb < K; kb += KT) {
        // cooperative, coalesced, divergence-free load of A tile [16 x KT]
        #pragma unroll
        for (int i = 0; i < (16 * KT) / NT; ++i) {
            const int idx = tid + i * NT;
            const int r = idx >> 6;              // / KT
            const int c = idx & (KT - 1);        // % KT
            sA[idx] = A[aoff0 + r * lda + kb + c];
        }
        __syncthreads();

        const float* lp = &sA[l16 * KT + 2 * half];
        #pragma unroll
        for (int kk = 0; kk < KT; kk += 4) {
            v2f a = *(const v2f*)lp;             // ds_load_b64 (paired by clang)
            lp += 4;
            v2f b;
            if (TRANSB) {
                b = *(const v2f*)(B + boff);     // global_load_b64, contiguous K
            } else {
                b.x = B[boff];                   // row k+2*half
                b.y = B[boff + ldb];             // row k+2*half+1
            }
            boff += bstep;
            acc = __builtin_amdgcn_wmma_f32_16x16x4_f32(
                      false, a, false, b, (short)0, acc, false, false);
        }
        __syncthreads();
    }

    #pragma unroll
    for (int i = 0; i < 8; ++i) {
        float v = acc[i];
        if (HASBIAS) v += bias[col];
        if (RELU)    v = fmaxf(v, 0.0f);
        C[(size_t)(row0 + i + 8 * half) * ldc + col] = v;
    }
}

// ---------------------------------------------------------------------------
// BatchNorm stats: biased mean/var over rows for each channel. Block/channel.
// ---------------------------------------------------------------------------
__global__ void bn_stats(const float* __restrict__ Z, int Mrows, int Ccols,
                         float* __restrict__ mean, float* __restrict__ var)
{
    const int c   = blockIdx.x;
    const int tid = threadIdx.x;
    float s = 0.0f, s2 = 0.0f;
    for (int r = tid; r < Mrows; r += blockDim.x) {
        float v = Z[(size_t)r * Ccols + c];
        s += v; s2 += v * v;
    }
    __shared__ float sh[256], sh2[256];
    sh[tid] = s; sh2[tid] = s2;
    __syncthreads();
    for (int off = blockDim.x >> 1; off > 0; off >>= 1) {
        if (tid < off) { sh[tid] += sh[tid + off]; sh2[tid] += sh2[tid + off]; }
        __syncthreads();
    }
    if (tid == 0) {
        float m = sh[0] / (float)Mrows;
        mean[c] = m;
        var[c]  = sh2[0] / (float)Mrows - m * m;
    }
}

// ---------------------------------------------------------------------------
// y = relu(g*(x-m)/sqrt(v+eps)+b); Z[e] = max over channels (relu>=0 so
// init 0 is exact). Row = 64 contiguous floats, 256B-aligned -> b128 loads.
// Writes into Z_H[:, zoff] (stride 2).
// ---------------------------------------------------------------------------
__global__ void bn_relu_max(const float* __restrict__ Zc,
                            const float* __restrict__ g, const float* __restrict__ b,
                            const float* __restrict__ mean, const float* __restrict__ var,
                            float* __restrict__ ZH, int zoff)
{
    __shared__ float scale[EH_], shift[EH_];
    const int tid = threadIdx.x;
    if (tid < EH_) {
        float rs = rsqrtf(var[tid] + BN_EPS);
        float sc = g[tid] * rs;
        scale[tid] = sc;
        shift[tid] = b[tid] - mean[tid] * sc;
    }
    __syncthreads();
    const int e = blockIdx.x * blockDim.x + tid;
    if (e < EE) {
        float mx = 0.0f;
        const v4f* row4 = (const v4f*)(Zc + (size_t)e * EH_);
        #pragma unroll
        for (int q = 0; q < EH_ / 4; ++q) {
            v4f v = row4[q];
            mx = fmaxf(mx, fmaxf(v.x * scale[4*q+0] + shift[4*q+0],
                                 v.y * scale[4*q+1] + shift[4*q+1]));
            mx = fmaxf(mx, fmaxf(v.z * scale[4*q+2] + shift[4*q+2],
                                 v.w * scale[4*q+3] + shift[4*q+3]));
        }
        ZH[(size_t)e * 2 + zoff] = fmaxf(mx, 0.0f);
    }
}

// ---------------------------------------------------------------------------
// Zt2[e,j] = Z1[e] * W2[j] + b2[j]   (rank-1: hosc layer-2 linear, in=1)
// ---------------------------------------------------------------------------
__global__ void outer_zt2(const float* __restrict__ ZH, const float* __restrict__ W2,
                          const float* __restrict__ b2, float* __restrict__ Zt2)
{
    const int i = blockIdx.x * blockDim.x + threadIdx.x;
    if (i < EE * EH_) {
        int e = i >> 6, j = i & 63;
        Zt2[i] = ZH[(size_t)e * 2 + 0] * W2[j] + b2[j];
    }
}

// ---------------------------------------------------------------------------
// H_e = B1 @ Z_H  -> Hcat[:,256:258].  One block per node row; coalesced
// float4 sweep over the 8192-wide B1 row (128MB stream -> b128 loads).
// ---------------------------------------------------------------------------
__global__ void fuse_edges(const float* __restrict__ B1, const float* __restrict__ ZH,
                           float* __restrict__ Hcat)
{
    const int i   = blockIdx.x;
    const int tid = threadIdx.x;
    const v4f* row4 = (const v4f*)(B1 + (size_t)i * EE);
    const v4f* zh4  = (const v4f*)ZH;
    float s0 = 0.0f, s1 = 0.0f;
    for (int q = tid; q < EE / 4; q += blockDim.x) {
        v4f b  = row4[q];
        v4f za = zh4[2 * q];        // Z0(e0),Z1(e0),Z0(e1),Z1(e1)
        v4f zb = zh4[2 * q + 1];    // Z0(e2),Z1(e2),Z0(e3),Z1(e3)
        s0 += b.x * za.x + b.y * za.z + b.z * zb.x + b.w * zb.z;
        s1 += b.x * za.y + b.y * za.w + b.z * zb.y + b.w * zb.w;
    }
    __shared__ float r0[256], r1[256];
    r0[tid] = s0; r1[tid] = s1;
    __syncthreads();
    for (int off = blockDim.x >> 1; off > 0; off >>= 1) {
        if (tid < off) { r0[tid] += r0[tid + off]; r1[tid] += r1[tid + off]; }
        __syncthreads();
    }
    if (tid == 0) {
        Hcat[(size_t)i * EMB_ + 256] = r0[0];
        Hcat[(size_t)i * EMB_ + 257] = r1[0];
    }
}

__device__ __forceinline__ float sigmoidf_(float x) {
    return 1.0f / (1.0f + expf(-x));
}

// ---------------------------------------------------------------------------
// node_prob[i] = sigmoid(Hcat[i,:] . W + b).  One wave32 per row.
// ---------------------------------------------------------------------------
__global__ void node_head(const float* __restrict__ Hcat, const float* __restrict__ W,
                          const float* __restrict__ b, float* __restrict__ out)
{
    const int gw   = (blockIdx.x * blockDim.x + threadIdx.x) >> 5;
    const int lane = threadIdx.x & 31;
    if (gw >= NN) return;
    const float* row = Hcat + (size_t)gw * EMB_;
    float s = 0.0f;
    for (int j = lane; j < EMB_; j += 32) s += row[j] * W[j];
    for (int m = 16; m > 0; m >>= 1) s += __shfl_xor(s, m, 32);
    if (lane == 0) out[gw] = sigmoidf_(s + b[0]);
}

// ---------------------------------------------------------------------------
// edge_prob[e] = sigmoid(ZH[e,0]*w0 + ZH[e,1]*w1 + b)
// ---------------------------------------------------------------------------
__global__ void edge_head(const float* __restrict__ ZH, const float* __restrict__ W,
                          const float* __restrict__ b, float* __restrict__ out)
{
    const int e = blockIdx.x * blockDim.x + threadIdx.x;
    if (e < EE)
        out[e] = sigmoidf_(ZH[(size_t)e * 2] * W[0] + ZH[(size_t)e * 2 + 1] * W[1] + b[0]);
}

// ---------------------------------------------------------------------------
// hyperedge_prob[h] = sigmoid(mean_k(Hcat[idx[h,k],:]) . W + b)
// dot is linear, so accumulate all K rows then divide. One wave32 per h.
// hyperedges are int64 in the reference.
// ---------------------------------------------------------------------------
__global__ void hyper_head(const float* __restrict__ Hcat,
                           const long long* __restrict__ idx,
                           const float* __restrict__ W, const float* __restrict__ b,
                           float* __restrict__ out)
{
    const int gw   = (blockIdx.x * blockDim.x + threadIdx.x) >> 5;
    const int lane = threadIdx.x & 31;
    if (gw >= HN_) return;
    float s = 0.0f;
    for (int k = 0; k < KK_; ++k) {
        const long long node = idx[(size_t)gw * KK_ + k];
        const float* row = Hcat + (size_t)node * EMB_;
        for (int j = lane; j < EMB_; j += 32) s += row[j] * W[j];
    }
    s *= (1.0f / (float)KK_);
    for (int m = 16; m > 0; m >>= 1) s += __shfl_xor(s, m, 32);
    if (lane == 0) out[gw] = sigmoidf_(s + b[0]);
}

// ---------------------------------------------------------------------------
// Launch
// ---------------------------------------------------------------------------
extern "C" void kernel_launch(void* const* d_in, const int* in_sizes, int n_in,
                              void* d_out, int out_size, void* d_ws, size_t ws_size,
                              hipStream_t stream)
{
    const float* X_n     = (const float*)d_in[0];
    const float* X_e     = (const float*)d_in[1];
    const float* A_tilde = (const float*)d_in[2];
    const float* L1      = (const float*)d_in[3];
    const float* B1      = (const float*)d_in[4];
    const float* gW1     = (const float*)d_in[5];
    const float* gb1     = (const float*)d_in[6];
    const float* gW2     = (const float*)d_in[7];
    const float* gb2     = (const float*)d_in[8];
    const float* hW1     = (const float*)d_in[9];
    const float* hb1     = (const float*)d_in[10];
    const float* bn1g    = (const float*)d_in[11];
    const float* bn1b    = (const float*)d_in[12];
    const float* hW2     = (const float*)d_in[13];
    const float* hb2     = (const float*)d_in[14];
    const float* bn2g    = (const float*)d_in[15];
    const float* bn2b    = (const float*)d_in[16];
    const float* nodeW   = (const float*)d_in[17];
    const float* nodeb   = (const float*)d_in[18];
    const float* edgeW   = (const float*)d_in[19];
    const float* edgeb   = (const float*)d_in[20];
    const float* hyperW  = (const float*)d_in[21];
    const float* hyperb  = (const float*)d_in[22];
    const long long* hedges = (const long long*)d_in[23];

    float* out       = (float*)d_out;
    float* node_prob = out;                       // [4096]
    float* edge_prob = out + NN;                  // [8192]
    float* hyper_prob= out + NN + EE;             // [1024]
    float* Hcat      = out + NN + EE + HN_;       // [4096,258]

    // workspace layout (floats)
    float* ws   = (float*)d_ws;
    float* AX   = ws;                             // 4096*128
    float* H    = AX  + (size_t)NN * NF_;         // 4096*256
    float* AH   = H   + (size_t)NN * NH1;         // 4096*256
    float* Zt   = AH  + (size_t)NN * NH2;         // 8192*64 (reused layer 2)
    float* Zc   = Zt  + (size_t)EE * EH_;         // 8192*64 (reused layer 2)
    float* ZH   = Zc  + (size_t)EE * EH_;         // 8192*2
    float* m1   = ZH  + (size_t)EE * 2;           // 64
    float* v1   = m1 + EH_;                       // 64
    float* m2   = v1 + EH_;                       // 64
    float* v2   = m2 + EH_;                       // 64
    (void)ws_size; (void)in_sizes; (void)n_in; (void)out_size;

    // --- GNN branch ---
    // AX = A_tilde @ X_n            [4096,128]
    gemm_wmma_f32<NF_/16, 0, 0, 0><<<NN/16, (NF_/16)*32, 0, stream>>>(
        A_tilde, NN, X_n, NF_, nullptr, AX, NF_, NN);
    // H = relu(AX @ W1^T + b1)      [4096,256]
    gemm_wmma_f32<NH1/16, 1, 1, 1><<<NN/16, (NH1/16)*32, 0, stream>>>(
        AX, NF_, gW1, NF_, gb1, H, NH1, NF_);
    // AH = A_tilde @ H              [4096,256]
    gemm_wmma_f32<NH1/16, 0, 0, 0><<<NN/16, (NH1/16)*32, 0, stream>>>(
        A_tilde, NN, H, NH1, nullptr, AH, NH1, NN);
    // H_n = relu(AH @ W2^T + b2) -> Hcat[:,0:256] (ldc = 258)
    gemm_wmma_f32<NH2/16, 1, 1, 1><<<NN/16, (NH2/16)*32, 0, stream>>>(
        AH, NH1, gW2, NH1, gb2, Hcat, EMB_, NH1);

    // --- HoSC layer 1 ---
    // Zt = X_e @ hW1^T + hb1        [8192,64]
    gemm_wmma_f32<EH_/16, 1, 1, 0><<<EE/16, (EH_/16)*32, 0, stream>>>(
        X_e, EF_, hW1, EF_, hb1, Zt, EH_, EF_);
    // Zc = L1 @ Zt                  [8192,64]
    gemm_wmma_f32<EH_/16, 0, 0, 0><<<EE/16, (EH_/16)*32, 0, stream>>>(
        L1, EE, Zt, EH_, nullptr, Zc, EH_, EE);
    bn_stats<<<EH_, 256, 0, stream>>>(Zc, EE, EH_, m1, v1);
    bn_relu_max<<<EE/256, 256, 0, stream>>>(Zc, bn1g, bn1b, m1, v1, ZH, 0);

    // --- HoSC layer 2 ---
    outer_zt2<<<(EE*EH_)/256, 256, 0, stream>>>(ZH, hW2, hb2, Zt);
    gemm_wmma_f32<EH_/16, 0, 0, 0><<<EE/16, (EH_/16)*32, 0, stream>>>(
        L1, EE, Zt, EH_, nullptr, Zc, EH_, EE);
    bn_stats<<<EH_, 256, 0, stream>>>(Zc, EE, EH_, m2, v2);
    bn_relu_max<<<EE/256, 256, 0, stream>>>(Zc, bn2g, bn2b, m2, v2, ZH, 1);

    // --- fuse + heads ---
    fuse_edges<<<NN, 256, 0, stream>>>(B1, ZH, Hcat);
    edge_head<<<EE/256, 256, 0, stream>>>(ZH, edgeW, edgeb, edge_prob);
    node_head<<<NN/8, 256, 0, stream>>>(Hcat, nodeW, nodeb, node_prob);
    hyper_head<<<HN_/8, 256, 0, stream>>>(Hcat, hedges, hyperW, hyperb, hyper_prob);
}